// PWCProbFlow_60859686584828
// MI455X (gfx1250) — compile-verified
//
#include <hip/hip_runtime.h>

typedef __attribute__((ext_vector_type(2))) float v2f;
typedef __attribute__((ext_vector_type(8))) float v8f;

#define BATCH 4
#define CH    32
#define H     192
#define W     448
#define NPIX  (CH * H * W)        // elements per image (one batch slice of one tensor)
#define PLANE (H * W)
#define TILE  16
#define XT    (W / TILE)          // 28 x-tiles

// ---------------------------------------------------------------------------
// Kernel 0: zero the 16 double accumulators in workspace
// ---------------------------------------------------------------------------
__global__ void init_ws_kernel(double* sums) {
    int i = threadIdx.x;
    if (i < 16) sums[i] = 0.0;
}

// ---------------------------------------------------------------------------
// Kernel 1: per-image sum / sumsq (double precision) with atomic f64 adds
// grid = (128, 4 /*b*/, 2 /*tensor*/), block = 256.  float4 (b128) loads.
// ---------------------------------------------------------------------------
__global__ void moments_kernel(const float* __restrict__ f1,
                               const float* __restrict__ f2,
                               double* __restrict__ sums) {
    const int b = blockIdx.y;
    const int t = blockIdx.z;
    const float4* base =
        (const float4*)(((t == 0) ? f1 : f2) + (size_t)b * NPIX);
    const int n4 = NPIX / 4;

    double s = 0.0, q = 0.0;
    for (int i = blockIdx.x * blockDim.x + threadIdx.x; i < n4;
         i += gridDim.x * blockDim.x) {
        float4 v = base[i];
        s += (double)v.x + (double)v.y + (double)v.z + (double)v.w;
        q += (double)v.x * v.x + (double)v.y * v.y +
             (double)v.z * v.z + (double)v.w * v.w;
    }

    __shared__ double ls[256];
    __shared__ double lq[256];
    ls[threadIdx.x] = s;
    lq[threadIdx.x] = q;
    __syncthreads();
    for (int off = 128; off > 0; off >>= 1) {
        if ((int)threadIdx.x < off) {
            ls[threadIdx.x] += ls[threadIdx.x + off];
            lq[threadIdx.x] += lq[threadIdx.x + off];
        }
        __syncthreads();
    }
    if (threadIdx.x == 0) {
        atomicAdd(&sums[(t * 4 + b) * 2 + 0], ls[0]);   // global_atomic_add_f64
        atomicAdd(&sums[(t * 4 + b) * 2 + 1], lq[0]);
    }
}

// ---------------------------------------------------------------------------
// Kernel 2: fold moments into per-b mean and 1/std (unbiased var ddof=1,
// per-image moments averaged across the two images, eps = 1e-16)
// ---------------------------------------------------------------------------
__global__ void stats_kernel(const double* __restrict__ sums,
                             float* __restrict__ mr) {
    int b = threadIdx.x;
    if (b < BATCH) {
        const double n  = (double)NPIX;
        double s1 = sums[(0 * 4 + b) * 2 + 0], q1 = sums[(0 * 4 + b) * 2 + 1];
        double s2 = sums[(1 * 4 + b) * 2 + 0], q2 = sums[(1 * 4 + b) * 2 + 1];
        double m1 = s1 / n, m2 = s2 / n;
        double v1 = (q1 - s1 * s1 / n) / (n - 1.0);
        double v2 = (q2 - s2 * s2 / n) / (n - 1.0);
        double mean = 0.5 * (m1 + m2);
        double rstd = 1.0 / sqrt(0.5 * (v1 + v2) + 1e-16);
        mr[b]         = (float)mean;
        mr[BATCH + b] = (float)rstd;
    }
}

// ---------------------------------------------------------------------------
// Kernel 3: WMMA cost volume.
// One wave32 per (b, y, x-tile). For each of 9 vertical shifts di, compute two
// 16x16 fp32 WMMA products D0/D1 (K=32 via 8 steps of v_wmma_f32_16x16x4_f32):
//   D[p, n] = sum_c f1n[c, x0+p] * f2n[c, (x0-4)+n]      (n 0..31 via 2 tiles)
// The band D[p, p+dj], dj=0..8, gives all 9 horizontal-shift costs for pixel
// x0+p.  Bounds are handled branch-free: addresses are clamped to legal
// locations and the normalization multiplier is zeroed for padded lanes
// (normalize-then-zero-pad semantics, identical to the reference).
// ---------------------------------------------------------------------------
__global__ void __launch_bounds__(32)
costvol_kernel(const float* __restrict__ f1,
               const float* __restrict__ f2,
               const float* __restrict__ mr,
               float* __restrict__ out) {
    const int x0   = blockIdx.x * TILE;   // 0..432
    const int y    = blockIdx.y;          // 0..191
    const int b    = blockIdx.z;          // 0..3
    const int lane = threadIdx.x;         // 0..31
    const int lo   = lane & 15;
    const int hi   = lane >> 4;

    const float mean = mr[b];
    const float rstd = mr[BATCH + b];

    const float* F1 = f1 + (size_t)b * NPIX;
    const float* F2 = f2 + (size_t)b * NPIX;

    // A-matrix (16x4 fp32 per K-step), ISA layout: VGPR0 = {K=0 | K=2},
    // VGPR1 = {K=1 | K=3}; lanes 0-15 / 16-31 hold M=0..15.
    v2f A[8];
#pragma unroll
    for (int k = 0; k < 8; ++k) {
        const int c0 = 4 * k + 2 * hi;
        const float* p0 = F1 + ((size_t)c0 * H + y) * W + x0 + lo;
        A[k].x = (p0[0]     - mean) * rstd;
        A[k].y = (p0[PLANE] - mean) * rstd;
    }

    // Branch-free bounds: clamp columns, zero the multiplier when padded.
    const int  col0  = x0 - 4  + lo;               // B tile 0 (may underflow)
    const int  col1  = x0 + 12 + lo;               // B tile 1 (may overflow)
    const int  col0c = (col0 < 0) ? 0 : col0;
    const int  col1c = (col1 >= W) ? (W - 1) : col1;
    const bool c0ok  = (col0 >= 0);
    const bool c1ok  = (col1 < W);

    // Per-lane channel-plane base offsets (32-bit element indices).
    unsigned base0[8], base1[8];
#pragma unroll
    for (int k = 0; k < 8; ++k) {
        const unsigned cb = (unsigned)(4 * k + 2 * hi) * PLANE;
        base0[k] = cb + (unsigned)col0c;
        base1[k] = cb + (unsigned)col1c;
    }

    __shared__ float lds[TILE][36];       // 16 x 32 (+pad) staging for D0|D1

    for (int di = 0; di < 9; ++di) {
        const int  yy    = y + di - 4;
        const bool rowok = (yy >= 0) && (yy < H);
        const int  yyc   = rowok ? yy : 0;
        const unsigned rowoff = (unsigned)yyc * W;

        const float w0 = (rowok && c0ok) ? rstd : 0.0f;
        const float w1 = (rowok && c1ok) ? rstd : 0.0f;

        // speculative prefetch of the next di row -> global_prefetch_b8
        {
            int yn = y + di - 3;
            if (yn >= 0 && yn < H)
                __builtin_prefetch(F2 + base0[0] + (unsigned)yn * W, 0, 0);
        }

        // -------- phase 1: issue all 32 loads for this row ----------------
        float r0x[8], r0y[8], r1x[8], r1y[8];
#pragma unroll
        for (int k = 0; k < 8; ++k) {
            r0x[k] = F2[base0[k] + rowoff];
            r0y[k] = F2[base0[k] + rowoff + PLANE];
            r1x[k] = F2[base1[k] + rowoff];
            r1y[k] = F2[base1[k] + rowoff + PLANE];
        }

        // -------- phase 2: normalize + 16 WMMAs ---------------------------
        v8f d0 = {};
        v8f d1 = {};
#pragma unroll
        for (int k = 0; k < 8; ++k) {
            v2f b0, b1;
            b0.x = (r0x[k] - mean) * w0;
            b0.y = (r0y[k] - mean) * w0;
            b1.x = (r1x[k] - mean) * w1;
            b1.y = (r1y[k] - mean) * w1;
            // 8 args: (neg_a, A, neg_b, B, c_mod, C, reuse_a, reuse_b)
            d0 = __builtin_amdgcn_wmma_f32_16x16x4_f32(false, A[k], false, b0,
                                                       (short)0, d0, false, false);
            d1 = __builtin_amdgcn_wmma_f32_16x16x4_f32(false, A[k], false, b1,
                                                       (short)0, d1, false, false);
        }

        // Stage D tiles: this lane holds rows M = r + 8*hi, column N = lo.
#pragma unroll
        for (int r = 0; r < 8; ++r) {
            lds[r + 8 * hi][lo]      = d0[r];
            lds[r + 8 * hi][16 + lo] = d1[r];
        }
        __syncthreads();

        // Emit the 144 band values: cost(p, dj) = D[p, p+dj] / 32.
        // Non-temporal stores keep the streamed 111MB output out of L2 so the
        // 88MB of inputs stay resident across their 18x reuse.
        const size_t obase = (((size_t)b * 81 + (size_t)di * 9) * H + y) * W + x0;
#pragma unroll
        for (int t = 0; t < 4; ++t) {
            int qidx = t * 32 + lane;
            int dj = qidx >> 4, p = qidx & 15;
            float v = lds[p][p + dj] * (1.0f / 32.0f);
            __builtin_nontemporal_store(v, &out[obase + (size_t)dj * PLANE + p]);
        }
        if (lane < 16) {
            int p = lane;
            float v = lds[p][p + 8] * (1.0f / 32.0f);
            __builtin_nontemporal_store(v, &out[obase + (size_t)8 * PLANE + p]);
        }
        __syncthreads();
    }
}

// ---------------------------------------------------------------------------
extern "C" void kernel_launch(void* const* d_in, const int* in_sizes, int n_in,
                              void* d_out, int out_size, void* d_ws, size_t ws_size,
                              hipStream_t stream) {
    const float* f1 = (const float*)d_in[0];
    const float* f2 = (const float*)d_in[1];
    float* out = (float*)d_out;

    double* sums = (double*)d_ws;                               // 16 doubles
    float*  mr   = (float*)((char*)d_ws + 16 * sizeof(double)); // mean[4], rstd[4]

    hipLaunchKernelGGL(init_ws_kernel, dim3(1), dim3(32), 0, stream, sums);
    hipLaunchKernelGGL(moments_kernel, dim3(128, BATCH, 2), dim3(256), 0, stream,
                       f1, f2, sums);
    hipLaunchKernelGGL(stats_kernel, dim3(1), dim3(BATCH), 0, stream, sums, mr);
    hipLaunchKernelGGL(costvol_kernel, dim3(XT, H, BATCH), dim3(32), 0, stream,
                       f1, f2, mr, out);
}